// Dinov2SelfAttention_21646635172650
// MI455X (gfx1250) — compile-verified
//
#include <hip/hip_runtime.h>
#include <math.h>

typedef float v2f __attribute__((ext_vector_type(2)));
typedef float v8f __attribute__((ext_vector_type(8)));
typedef int   v4i __attribute__((ext_vector_type(4)));

// D = A(16x4) * B(4x16) + C, fp32 (only fp32 WMMA on CDNA5)
#define WMMA_F32(a, b, c)                                                     \
  __builtin_amdgcn_wmma_f32_16x16x4_f32(false, (a), false, (b), (short)0,     \
                                        (c), false, false)

// ---- async global->LDS (CDNA5), with synchronous fallback ------------------
#if defined(__has_builtin)
#if __has_builtin(__builtin_amdgcn_global_load_async_to_lds_b128)
#define HAVE_ASYNC_LDS 1
#endif
#endif

#ifdef HAVE_ASYNC_LDS
typedef __attribute__((address_space(1))) v4i* gv4i_p;  // global src
typedef __attribute__((address_space(3))) v4i* lv4i_p;  // LDS dst
#define CP16(g, l)                                                            \
  __builtin_amdgcn_global_load_async_to_lds_b128((gv4i_p)(g), (lv4i_p)(l), 0, 0)
#define ASYNC_WAIT0() asm volatile("s_wait_asynccnt 0x0" ::: "memory")
#else
#define CP16(g, l) (*(float4*)(l) = *(const float4*)(g))
#define ASYNC_WAIT0() ((void)0)
#endif

constexpr int Bn  = 8;
constexpr int Sn  = 1370;
constexpr int Dn  = 1024;
constexpr int Hn  = 16;
constexpr int HDn = 64;
constexpr int NTOT = Bn * Sn;  // 10960 rows

// ---------------------------------------------------------------------------
// Kernel 1: fused QKV projection.  out[b,h,s,hd] = X[b,s,:] . W[o,:] + bias[o]
// 256 threads = 8 waves, 128x128 tile, K chunks of 32 double-buffered in LDS.
// OOB rows are CLAMPED (not zeroed): WMMA row m only affects output row m,
// and clamped rows are never stored -> unconditional b128 staging.
// ---------------------------------------------------------------------------
__device__ __forceinline__ void k1_stage(const float* __restrict__ X,
                                         const float* __restrict__ W,
                                         float* xbuf, float* wbuf,
                                         int rowBase, int colBase, int kc) {
#pragma unroll
  for (int i = 0; i < 4; ++i) {
    int idx = threadIdx.x + i * 256;  // 0..1023 float4 slots per tile
    int row = idx >> 3;
    int c4  = (idx & 7) << 2;
    int rg  = rowBase + row;
    if (rg > NTOT - 1) rg = NTOT - 1;  // clamp, see note above
    CP16(X + (size_t)rg * Dn + kc + c4, xbuf + row * 36 + c4);
    CP16(W + (size_t)(colBase + row) * Dn + kc + c4, wbuf + row * 36 + c4);
  }
}

__global__ __launch_bounds__(256) void qkv_kernel(
    const float* __restrict__ X,
    const float* __restrict__ Wq, const float* __restrict__ bq,
    const float* __restrict__ Wk, const float* __restrict__ bk,
    const float* __restrict__ Wv, const float* __restrict__ bv,
    float* __restrict__ Qo, float* __restrict__ Ko, float* __restrict__ Vo) {
  __shared__ float Xs[2][128 * 36];  // 128 rows x 32 k, stride 36 (pad)
  __shared__ float Ws[2][128 * 36];

  const float* W;
  const float* bias;
  float* out;
  if (blockIdx.z == 0) { W = Wq; bias = bq; out = Qo; }
  else if (blockIdx.z == 1) { W = Wk; bias = bk; out = Ko; }
  else { W = Wv; bias = bv; out = Vo; }

  const int lane = threadIdx.x & 31;
  const int wv   = threadIdx.x >> 5;   // 0..7
  const int wrow = wv >> 1;            // 0..3  (32 rows each)
  const int wcol = wv & 1;             // 0..1  (64 cols each)
  const int rowBase = blockIdx.y * 128;
  const int colBase = blockIdx.x * 128;

  const int m  = lane & 15;
  const int kh = 2 * (lane >> 4);      // 0 or 2 (K pair select)

  v8f acc[2][4];
#pragma unroll
  for (int rt = 0; rt < 2; ++rt)
#pragma unroll
    for (int ct = 0; ct < 4; ++ct) acc[rt][ct] = (v8f){0, 0, 0, 0, 0, 0, 0, 0};

  constexpr int NCHUNK = Dn / 32;  // 32
  k1_stage(X, W, Xs[0], Ws[0], rowBase, colBase, 0);

  for (int ci = 0; ci < NCHUNK; ++ci) {
    ASYNC_WAIT0();       // chunk ci resident in LDS (this wave's copies)
    __syncthreads();     // all waves done with buf[(ci+1)&1] (read at ci-1)
    if (ci + 1 < NCHUNK)
      k1_stage(X, W, Xs[(ci + 1) & 1], Ws[(ci + 1) & 1], rowBase, colBase,
               (ci + 1) * 32);  // overlaps with compute below

    const float* xb = Xs[ci & 1];
    const float* wb = Ws[ci & 1];
#pragma unroll
    for (int kk = 0; kk < 32; kk += 4) {
      v2f a[2], b[4];
#pragma unroll
      for (int rt = 0; rt < 2; ++rt) {
        const float* p = xb + (wrow * 32 + rt * 16 + m) * 36 + kk + kh;
        a[rt][0] = p[0]; a[rt][1] = p[1];
      }
#pragma unroll
      for (int ct = 0; ct < 4; ++ct) {
        const float* p = wb + (wcol * 64 + ct * 16 + m) * 36 + kk + kh;
        b[ct][0] = p[0]; b[ct][1] = p[1];
      }
#pragma unroll
      for (int rt = 0; rt < 2; ++rt)
#pragma unroll
        for (int ct = 0; ct < 4; ++ct)
          acc[rt][ct] = WMMA_F32(a[rt], b[ct], acc[rt][ct]);
    }
  }

  // bias + scatter to [B,H,S,hd]
#pragma unroll
  for (int rt = 0; rt < 2; ++rt)
#pragma unroll
    for (int ct = 0; ct < 4; ++ct) {
      int colg = colBase + wcol * 64 + ct * 16 + m;
      int h  = colg >> 6;
      int hd = colg & 63;
      float bvv = bias[colg];
#pragma unroll
      for (int r = 0; r < 8; ++r) {
        int rowg = rowBase + wrow * 32 + rt * 16 + r + 8 * (lane >> 4);
        if (rowg < NTOT) {
          int bb = rowg / Sn;
          int ss = rowg - bb * Sn;
          out[(((size_t)(bb * Hn + h)) * Sn + ss) * HDn + hd] =
              acc[rt][ct][r] + bvv;
        }
      }
    }
}

// ---------------------------------------------------------------------------
// Kernel 2: flash attention, fp32 WMMA.  128 threads = 4 waves; block owns 64
// query rows (16 per wave); 16-key blocks double-buffered with async copies.
// Tail keys: clamped loads give duplicate rows, but score columns >= S are
// masked to -1e30 (K side) and p==0 kills the V side.
// ---------------------------------------------------------------------------
__device__ __forceinline__ void k2_stage(const float* __restrict__ Kp,
                                         const float* __restrict__ Vp,
                                         float* kbuf, float* vbuf, int kb) {
#pragma unroll
  for (int i = 0; i < 2; ++i) {
    int idx = threadIdx.x + i * 128;  // 0..255 float4 slots per tile
    int row = idx >> 4;
    int c4  = (idx & 15) << 2;
    int kg  = kb * 16 + row;
    if (kg > Sn - 1) kg = Sn - 1;  // clamp
    CP16(Kp + (size_t)kg * HDn + c4, kbuf + row * 68 + c4);
    CP16(Vp + (size_t)kg * HDn + c4, vbuf + row * 68 + c4);
  }
}

__global__ __launch_bounds__(128) void attn_kernel(
    const float* __restrict__ Q, const float* __restrict__ K,
    const float* __restrict__ V, float* __restrict__ out) {
  __shared__ float Qs[64 * 68];
  __shared__ float Ks[2][16 * 68];
  __shared__ float Vs[2][16 * 68];
  __shared__ float Ps[4 * 16 * 20];  // per-wave P tile (C->A relayout)

  const int bh = blockIdx.y;
  const int bb = bh >> 4;
  const int hh = bh & 15;
  const int qbase = blockIdx.x * 64;
  const float* Qp = Q + (size_t)bh * Sn * HDn;
  const float* Kp = K + (size_t)bh * Sn * HDn;
  const float* Vp = V + (size_t)bh * Sn * HDn;

  const int lane = threadIdx.x & 31;
  const int wv   = threadIdx.x >> 5;  // 0..3
  const int m    = lane & 15;
  const int kh   = 2 * (lane >> 4);

  // stage Q (pre-scaled by 1/sqrt(64) = 0.125); clamped rows never stored
#pragma unroll
  for (int i = 0; i < 8; ++i) {
    int idx = threadIdx.x + i * 128;   // 0..1023 float4s
    int row = idx >> 4;
    int c4  = (idx & 15) << 2;
    int qg  = qbase + row;
    if (qg > Sn - 1) qg = Sn - 1;
    float4 qv = *(const float4*)(Qp + (size_t)qg * HDn + c4);
    qv.x *= 0.125f; qv.y *= 0.125f; qv.z *= 0.125f; qv.w *= 0.125f;
    *(float4*)(Qs + row * 68 + c4) = qv;
  }
  __syncthreads();

  // Q A-fragments in registers: 16 chunks of 16x4
  v2f qa[16];
  {
    const float* qr = Qs + (wv * 16 + m) * 68;
#pragma unroll
    for (int c = 0; c < 16; ++c) {
      qa[c][0] = qr[c * 4 + kh];
      qa[c][1] = qr[c * 4 + kh + 1];
    }
  }

  v8f acc[4];
#pragma unroll
  for (int t = 0; t < 4; ++t) acc[t] = (v8f){0, 0, 0, 0, 0, 0, 0, 0};
  float mrow[8], lrow[8];
#pragma unroll
  for (int r = 0; r < 8; ++r) { mrow[r] = -1e30f; lrow[r] = 0.f; }

  const int nkb = (Sn + 15) / 16;  // 86
  k2_stage(Kp, Vp, Ks[0], Vs[0], 0);

  for (int kb = 0; kb < nkb; ++kb) {
    ASYNC_WAIT0();
    __syncthreads();
    if (kb + 1 < nkb)
      k2_stage(Kp, Vp, Ks[(kb + 1) & 1], Vs[(kb + 1) & 1], kb + 1);

    const float* Kc = Ks[kb & 1];
    const float* Vc = Vs[kb & 1];

    // scores S = Q . K^T  (16 wmma, K-dim = 64)
    v8f s = (v8f){0, 0, 0, 0, 0, 0, 0, 0};
#pragma unroll
    for (int c = 0; c < 16; ++c) {
      v2f kbv;
      kbv[0] = Kc[m * 68 + c * 4 + kh];
      kbv[1] = Kc[m * 68 + c * 4 + kh + 1];
      s = WMMA_F32(qa[c], kbv, s);
    }

    const bool kvalid = (kb * 16 + m) < Sn;  // this lane's key column

    // online softmax per owned row; xor masks 1/2/4/8 stay inside each
    // 16-lane half, which is exactly one row of the 16x16 C layout.
#pragma unroll
    for (int r = 0; r < 8; ++r) {
      float sv = kvalid ? s[r] : -1e30f;
      float bm = sv;
      bm = fmaxf(bm, __shfl_xor(bm, 1));
      bm = fmaxf(bm, __shfl_xor(bm, 2));
      bm = fmaxf(bm, __shfl_xor(bm, 4));
      bm = fmaxf(bm, __shfl_xor(bm, 8));
      float mnew  = fmaxf(mrow[r], bm);
      float alpha = __expf(mrow[r] - mnew);
      float p     = __expf(sv - mnew);
      float ps = p;
      ps += __shfl_xor(ps, 1);
      ps += __shfl_xor(ps, 2);
      ps += __shfl_xor(ps, 4);
      ps += __shfl_xor(ps, 8);
      lrow[r] = lrow[r] * alpha + ps;
      mrow[r] = mnew;
#pragma unroll
      for (int t = 0; t < 4; ++t) acc[t][r] *= alpha;
      Ps[wv * 320 + (r + 8 * (lane >> 4)) * 20 + m] = p;  // C->A via LDS
    }

    // ctx += P(16x16) . V(16x64)  (16 wmma); same-wave LDS RAW -> dscnt waits
#pragma unroll
    for (int c = 0; c < 4; ++c) {
      v2f pa;
      pa[0] = Ps[wv * 320 + m * 20 + c * 4 + kh];
      pa[1] = Ps[wv * 320 + m * 20 + c * 4 + kh + 1];
#pragma unroll
      for (int t = 0; t < 4; ++t) {
        v2f vbv;
        vbv[0] = Vc[(c * 4 + kh) * 68 + t * 16 + m];
        vbv[1] = Vc[(c * 4 + kh + 1) * 68 + t * 16 + m];
        acc[t] = WMMA_F32(pa, vbv, acc[t]);
      }
    }
  }

  // normalize and store: out[b, s, h*64+hd]
#pragma unroll
  for (int t = 0; t < 4; ++t)
#pragma unroll
    for (int r = 0; r < 8; ++r) {
      int qg = qbase + wv * 16 + r + 8 * (lane >> 4);
      if (qg < Sn)
        out[((size_t)bb * Sn + qg) * Dn + hh * 64 + t * 16 + m] =
            acc[t][r] / lrow[r];
    }
}

// ---------------------------------------------------------------------------
extern "C" void kernel_launch(void* const* d_in, const int* in_sizes, int n_in,
                              void* d_out, int out_size, void* d_ws,
                              size_t ws_size, hipStream_t stream) {
  (void)in_sizes; (void)n_in; (void)out_size; (void)ws_size;
  const float* X  = (const float*)d_in[0];
  const float* Wq = (const float*)d_in[1];
  const float* bq = (const float*)d_in[2];
  const float* Wk = (const float*)d_in[3];
  const float* bk = (const float*)d_in[4];
  const float* Wv = (const float*)d_in[5];
  const float* bv = (const float*)d_in[6];
  float* out = (float*)d_out;

  size_t per = (size_t)Bn * Hn * Sn * HDn;  // 11,223,040 floats
  float* Qw = (float*)d_ws;
  float* Kw = Qw + per;
  float* Vw = Kw + per;

  dim3 g1(Dn / 128, (NTOT + 127) / 128, 3);   // 8 x 86 x 3
  qkv_kernel<<<g1, 256, 0, stream>>>(X, Wq, bq, Wk, bk, Wv, bv, Qw, Kw, Vw);

  dim3 g2((Sn + 63) / 64, Bn * Hn);           // 22 x 128
  attn_kernel<<<g2, 128, 0, stream>>>(Qw, Kw, Vw, out);
}